// Diffusion_Model_76510547411612
// MI455X (gfx1250) — compile-verified
//
#include <hip/hip_runtime.h>
#include <math.h>

typedef float v2f __attribute__((ext_vector_type(2)));
typedef float v8f __attribute__((ext_vector_type(8)));

#define BB   4096
#define TT   32
#define NN   200
#define LDX  201          // x last-dim length (N+1)
#define XBS  (TT * LDX)   // per-batch stride = 6432
#define ROWS (BB * NN)    // 819200 (b,node) rows
#define TILES_PER_BLOCK 8
#define ROWS_PER_TILE   16

// integer power by squaring, e in [0,31]; exact sign handling for negative base
__device__ __forceinline__ float powi5(float base, int e) {
    float r = 1.f, b = base;
#pragma unroll
    for (int i = 0; i < 5; ++i) {
        if (e & 1) r *= b;
        b *= b;
        e >>= 1;
    }
    return r;
}

// ---------------------------------------------------------------------------
// Kernel 0: fold (x - mean)/std into the first-layer weights:
//   W' [t,k] = W[t,k]/std[t],   b'[k] = b[k] - sum_t mean[t]*W[t,k]/std[t]
// ---------------------------------------------------------------------------
__global__ void prep_kernel(
    const float* __restrict__ W11, const float* __restrict__ b11,
    const float* __restrict__ W21, const float* __restrict__ b21,
    const float* __restrict__ xmean, const float* __restrict__ xstd,
    float* __restrict__ W11s, float* __restrict__ b11s,
    float* __restrict__ W21s, float* __restrict__ b21s)
{
    const int k = threadIdx.x;
    if (k >= 32) return;
    float acc1 = b11[k], acc2 = b21[k];
#pragma unroll 4
    for (int t = 0; t < TT; ++t) {
        const float m = xmean[t], s = xstd[t];
        const float w1 = W11[t * 32 + k] / s;
        const float w2 = W21[t * 32 + k] / s;
        W11s[t * 32 + k] = w1;
        W21s[t * 32 + k] = w2;
        acc1 = fmaf(-m, w1, acc1);
        acc2 = fmaf(-m, w2, acc2);
    }
    b11s[k] = acc1;
    b21s[k] = acc2;
}

// ---------------------------------------------------------------------------
// Kernel 1: xd[b] = sigmoid(relu(down_raw @ W21' + b21') @ W22 + b22)
// (scaling folded into W21'/b21'); depends only on b.
// ---------------------------------------------------------------------------
__global__ void __launch_bounds__(256) xd_kernel(
    const float* __restrict__ x,
    const float* __restrict__ W21s, const float* __restrict__ b21s,
    const float* __restrict__ W22, const float* __restrict__ b22,
    float* __restrict__ xd_out)
{
    const int b = blockIdx.x * blockDim.x + threadIdx.x;
    if (b >= BB) return;
    float dn[TT];
#pragma unroll
    for (int t = 0; t < TT; ++t)
        dn[t] = x[(size_t)b * XBS + (size_t)t * LDX + NN];
    float s = b22[0];
#pragma unroll 4
    for (int k = 0; k < 32; ++k) {
        float acc = b21s[k];
#pragma unroll
        for (int t = 0; t < TT; ++t) acc = fmaf(dn[t], W21s[t * 32 + k], acc);
        s = fmaf(fmaxf(acc, 0.f), W22[k], s);
    }
    xd_out[b] = 1.f / (1.f + expf(-s));
}

// ---------------------------------------------------------------------------
// Kernel 2: per-wave tile of 16 (b,node) rows.
//   - async-load raw up[16][32] directly to LDS (ASYNCcnt path, wave-private)
//   - WMMA f32 16x16x4 chain: H = up_raw @ W11' (two 16-col halves)
//   - relu + @W12 + shfl-tree -> per-row scalar -> sigmoid -> v, Tt, F
//   - geometric decay dot with raw up (2 lanes/row, carried power, no powf)
// ---------------------------------------------------------------------------
__global__ void __launch_bounds__(256) main_kernel(
    const float* __restrict__ x,
    const float* __restrict__ W11s, const float* __restrict__ b11s,
    const float* __restrict__ W12, const float* __restrict__ b12,
    const float* __restrict__ W3,  const float* __restrict__ b3,
    const float* __restrict__ xd_arr,
    float* __restrict__ rowsum)
{
    __shared__ float tile[TILES_PER_BLOCK][ROWS_PER_TILE][TT + 1];
    const int lane = threadIdx.x & 31;
    const int w    = threadIdx.x >> 5;
    const int tile_row0 = (blockIdx.x * TILES_PER_BLOCK + w) * ROWS_PER_TILE;

    // ---- async global->LDS: lane (r = lane&15) loads one row, j parity (even/odd) by lane>>4
    {
        const int r    = lane & 15;
        const int grow = tile_row0 + r;
        const int br   = grow / NN;
        const int node = grow - br * NN;
        const unsigned voff0 = (unsigned)(br * XBS + node) * 4u;
        const int jh   = lane >> 4;                       // 0 or 1
        const unsigned lds0 = (unsigned)(uintptr_t)(&tile[w][r][0]);
        const unsigned long long xaddr = (unsigned long long)(uintptr_t)x;
#pragma unroll
        for (int jj = 0; jj < 16; ++jj) {
            const int j = jj * 2 + jh;
            const unsigned voff  = voff0 + (unsigned)(j * (LDX * 4));
            const unsigned laddr = lds0 + (unsigned)(j * 4);
            asm volatile("global_load_async_to_lds_b32 %0, %1, %2"
                         :: "v"(laddr), "v"(voff), "s"(xaddr) : "memory");
        }
        asm volatile("s_wait_asynccnt 0" ::: "memory");
    }
    // tile slice is wave-private: no workgroup barrier needed.

    // ---- WMMA: A = raw up (16x32 f32), B = W11' (32x32), K chained 8x4
    const int m    = lane & 15;                 // A row / B,C col
    const int koff = (lane >> 4) << 1;          // 0 or 2 (32-bit A/B K layout)
    v8f acc0 = {0.f, 0.f, 0.f, 0.f, 0.f, 0.f, 0.f, 0.f};
    v8f acc1 = {0.f, 0.f, 0.f, 0.f, 0.f, 0.f, 0.f, 0.f};
#pragma unroll
    for (int kb = 0; kb < 8; ++kb) {
        const int k0 = kb * 4 + koff;
        v2f A, B0, B1;
        A.x = tile[w][m][k0];
        A.y = tile[w][m][k0 + 1];
        B0.x = W11s[k0 * 32 + m];        B0.y = W11s[(k0 + 1) * 32 + m];
        B1.x = W11s[k0 * 32 + 16 + m];   B1.y = W11s[(k0 + 1) * 32 + 16 + m];
        acc0 = __builtin_amdgcn_wmma_f32_16x16x4_f32(false, A, false, B0,
                                                     (short)0, acc0, false, false);
        acc1 = __builtin_amdgcn_wmma_f32_16x16x4_f32(false, A, false, B1,
                                                     (short)0, acc1, false, false);
    }

    // ---- relu + second layer + 16-lane reduction (C layout: VGPR i -> row i / i+8)
    const float bia0 = b11s[m],     bia1 = b11s[16 + m];
    const float w120 = W12[m],      w121 = W12[16 + m];
    float p[8];
#pragma unroll
    for (int i = 0; i < 8; ++i) {
        const float h0 = fmaxf(acc0[i] + bia0, 0.f);
        const float h1 = fmaxf(acc1[i] + bia1, 0.f);
        float pv = fmaf(h0, w120, h1 * w121);
        pv += __shfl_xor(pv, 1, 32);
        pv += __shfl_xor(pv, 2, 32);
        pv += __shfl_xor(pv, 4, 32);
        pv += __shfl_xor(pv, 8, 32);
        p[i] = pv;
    }
    // lane picks its row scalar: rows 0-7 in lanes 0-15, rows 8-15 in lanes 16-31
    const int isel = lane & 7;
    float s = p[0];
#pragma unroll
    for (int i = 1; i < 8; ++i) s = (isel == i) ? p[i] : s;

    const int rloc = isel + ((lane >> 4) << 3);   // row within tile, 0..15
    const int grow = tile_row0 + rloc;
    const int br   = grow / NN;

    const float xu = 1.f / (1.f + expf(-(s + b12[0])));
    const float xd = xd_arr[br];
    const float v  = fmaf(xu, W3[0], fmaf(xd, W3[1], b3[0]));
    const float Tt = 45.0f / v + 1e-5f;
    float Tidx = rintf(Tt * 0.1f);                // round half-to-even, like jnp.round
    Tidx = (Tidx < 0.f)  ? 0.f  : Tidx;
    Tidx = (Tidx > 32.f) ? 31.f : Tidx;
    const float nlen = fmaxf(32.f - Tidx, 1.f);
    const float F    = 1.f / (1.f + Tt);          // ALPHA = 1
    const float omF  = 1.f - F;

    // two lanes per row split j in [0,16) and [16,32); geometric carry, no powf.
    // jta = last valid j (nlen-1); (int)NaN -> 0 => jta=-1 => zero contribution,
    // matching the reference's masked-NaN behavior.
    const int jstart = ((lane >> 3) & 1) << 4;    // 0 or 16
    const int jta = (int)nlen - 1;
    int e0 = jta - (jstart + 15);
    if (e0 < 0) e0 = 0;
    float g = F * powi5(omF, e0);                 // weight at lane's top active j
    float dec = 0.f;
#pragma unroll
    for (int jj = 15; jj >= 0; --jj) {
        const int j = jstart + jj;
        if (j <= jta) {
            dec = fmaf(g, tile[w][rloc][j], dec);
            g *= omF;
        }
    }
    dec += __shfl_xor(dec, 8, 32);
    if (jstart == 0) rowsum[grow] = dec;
}

// ---------------------------------------------------------------------------
// Kernel 3: deterministic fixed-tree reduction over the 200 nodes of each b.
// ---------------------------------------------------------------------------
__global__ void __launch_bounds__(256) reduce_kernel(
    const float* __restrict__ rowsum, float* __restrict__ out)
{
    __shared__ float red[256];
    const int b = blockIdx.x;
    const int t = threadIdx.x;
    red[t] = (t < NN) ? rowsum[(size_t)b * NN + t] : 0.f;
    __syncthreads();
#pragma unroll
    for (int sft = 128; sft > 0; sft >>= 1) {
        if (t < sft) red[t] += red[t + sft];
        __syncthreads();
    }
    if (t == 0) out[b] = red[0];
}

// ---------------------------------------------------------------------------
extern "C" void kernel_launch(void* const* d_in, const int* in_sizes, int n_in,
                              void* d_out, int out_size, void* d_ws, size_t ws_size,
                              hipStream_t stream) {
    (void)in_sizes; (void)n_in; (void)out_size; (void)ws_size;
    const float* x     = (const float*)d_in[0];
    const float* W11   = (const float*)d_in[1];
    const float* b11   = (const float*)d_in[2];
    const float* W12   = (const float*)d_in[3];
    const float* b12   = (const float*)d_in[4];
    const float* W21   = (const float*)d_in[5];
    const float* b21   = (const float*)d_in[6];
    const float* W22   = (const float*)d_in[7];
    const float* b22   = (const float*)d_in[8];
    const float* W3    = (const float*)d_in[9];
    const float* b3    = (const float*)d_in[10];
    const float* xmean = (const float*)d_in[11];
    const float* xstd  = (const float*)d_in[12];
    float* out    = (float*)d_out;

    float* xd_ws  = (float*)d_ws;          // BB floats
    float* rowsum = xd_ws + BB;            // ROWS floats
    float* W11s   = rowsum + ROWS;         // 1024
    float* b11s   = W11s + 1024;           // 32
    float* W21s   = b11s + 32;             // 1024
    float* b21s   = W21s + 1024;           // 32

    hipLaunchKernelGGL(prep_kernel, dim3(1), dim3(32), 0, stream,
                       W11, b11, W21, b21, xmean, xstd, W11s, b11s, W21s, b21s);

    hipLaunchKernelGGL(xd_kernel, dim3(BB / 256), dim3(256), 0, stream,
                       x, W21s, b21s, W22, b22, xd_ws);

    const int nblocks = ROWS / (ROWS_PER_TILE * TILES_PER_BLOCK);  // 6400
    hipLaunchKernelGGL(main_kernel, dim3(nblocks), dim3(256), 0, stream,
                       x, W11s, b11s, W12, b12, W3, b3, xd_ws, rowsum);

    hipLaunchKernelGGL(reduce_kernel, dim3(BB), dim3(256), 0, stream, rowsum, out);
}